// BatchHardTripletLoss_87471303950746
// MI455X (gfx1250) — compile-verified
//
#include <hip/hip_runtime.h>
#include <math.h>

typedef float v2f __attribute__((ext_vector_type(2)));
typedef float v8f __attribute__((ext_vector_type(8)));

#define N_ROWS 8192
#define DIM    128
#define TMARGIN 0.3f
#define CSPLIT 4
#define COLS_PER_SPLIT (N_ROWS / CSPLIT)   // 2048
#define BLK_COLS 64
#define LDS_STRIDE 130                     // 128 + 2 pad -> conflict-free f32x2 reads

// workspace layout (in floats)
#define WS_SQ 0
#define WS_HP (N_ROWS)
#define WS_HN (N_ROWS + CSPLIT * N_ROWS)

// ---------------------------------------------------------------------------
// Kernel 1: per-row squared norms. One wave32 per row: 4 floats/lane + shfl.
// ---------------------------------------------------------------------------
__global__ __launch_bounds__(256) void bh_row_sqnorm(const float* __restrict__ feats,
                                                     float* __restrict__ sq) {
    const int wave = threadIdx.x >> 5;
    const int lane = threadIdx.x & 31;
    const int row  = blockIdx.x * 8 + wave;
    const float4 v = ((const float4*)(feats + row * DIM))[lane];
    float s = v.x * v.x + v.y * v.y + v.z * v.z + v.w * v.w;
#pragma unroll
    for (int off = 16; off > 0; off >>= 1) s += __shfl_xor(s, off, 32);
    if (lane == 0) sq[row] = s;
}

// ---------------------------------------------------------------------------
// Kernel 2: fused Gram-tile (fp32 WMMA 16x16x4) + hardest pos/neg on SQUARED
// distances. Block = 256 threads (8 waves). Wave w owns rows [bx*128+w*16,+16).
// blockIdx.y = column split (2048 cols), staged via LDS in 64-col blocks.
// The 4 sub-tiles of each block are computed in ONE fused K-loop -> 4
// independent WMMA accumulation chains (hides matrix-pipe latency).
// Per element we track t = csq - 2*dot; rowsq added at write-out; clamp/eps/
// sqrt deferred to finalize (all monotone). Diag test in a rare uniform branch.
// ---------------------------------------------------------------------------
__global__ __launch_bounds__(256) void bh_tile_kernel(const float* __restrict__ feats,
                                                      const int*   __restrict__ labels,
                                                      const float* __restrict__ sq,
                                                      float* __restrict__ hp_part,
                                                      float* __restrict__ hn_part) {
    __shared__ float smem[BLK_COLS * LDS_STRIDE];

    const int lane = threadIdx.x & 31;
    const int wave = threadIdx.x >> 5;
    const int half = lane >> 4;         // 0: K+0/K+1 , 1: K+2/K+3 (fp32 WMMA layout)
    const int l15  = lane & 15;
    const int i0   = blockIdx.x * 128 + wave * 16;
    const int cs   = blockIdx.y;
    const int col0 = cs * COLS_PER_SPLIT;

    // Preload this wave's A fragments: 16 rows x K=128 -> 32 k-steps of float2/lane.
    const int arow = i0 + l15;
    v2f afrag[32];
#pragma unroll
    for (int kk = 0; kk < 32; ++kk)
        afrag[kk] = *(const v2f*)(feats + arow * DIM + kk * 4 + half * 2);

    // Row-side labels for the 8 accumulator slots this lane produces.
    int rowlab[8];
#pragma unroll
    for (int p = 0; p < 8; ++p) rowlab[p] = labels[i0 + p + half * 8];

    float hp[8], hn[8];   // running max/min of (csq - 2*dot)
#pragma unroll
    for (int p = 0; p < 8; ++p) { hp[p] = -__builtin_inff(); hn[p] = __builtin_inff(); }

    const int nblk = COLS_PER_SPLIT / BLK_COLS;
    for (int blk = 0; blk < nblk; ++blk) {
        const int cb0 = col0 + blk * BLK_COLS;
        __syncthreads();
        // Stage 64 feature rows (the "columns") into padded LDS, float2 copies.
        {
            const v2f* src = (const v2f*)(feats + (size_t)cb0 * DIM);
            for (int idx = threadIdx.x; idx < BLK_COLS * (DIM / 2); idx += 256) {
                const int r  = idx >> 6;      // DIM/2 == 64
                const int c2 = idx & 63;
                *(v2f*)(&smem[r * LDS_STRIDE + c2 * 2]) = src[r * 64 + c2];
            }
            // Warm WGP cache for the next column block (128B slice per thread).
            if (blk + 1 < nblk) {
                const char* nxt = (const char*)(feats + (size_t)(cb0 + BLK_COLS) * DIM);
                __builtin_prefetch(nxt + threadIdx.x * 128, 0, 3);
            }
        }
        __syncthreads();

        // ---- fused K-loop over the 4 sub-tiles: 4 independent WMMA chains ----
        const float* bb0 = &smem[(0 * 16 + l15) * LDS_STRIDE + half * 2];
        const float* bb1 = &smem[(1 * 16 + l15) * LDS_STRIDE + half * 2];
        const float* bb2 = &smem[(2 * 16 + l15) * LDS_STRIDE + half * 2];
        const float* bb3 = &smem[(3 * 16 + l15) * LDS_STRIDE + half * 2];
        v8f acc0 = {}, acc1 = {}, acc2 = {}, acc3 = {};
#pragma unroll
        for (int kk = 0; kk < 32; ++kk) {
            const v2f a  = afrag[kk];
            const v2f b0 = *(const v2f*)(bb0 + kk * 4);
            const v2f b1 = *(const v2f*)(bb1 + kk * 4);
            const v2f b2 = *(const v2f*)(bb2 + kk * 4);
            const v2f b3 = *(const v2f*)(bb3 + kk * 4);
            acc0 = __builtin_amdgcn_wmma_f32_16x16x4_f32(false, a, false, b0, (short)0, acc0, false, false);
            acc1 = __builtin_amdgcn_wmma_f32_16x16x4_f32(false, a, false, b1, (short)0, acc1, false, false);
            acc2 = __builtin_amdgcn_wmma_f32_16x16x4_f32(false, a, false, b2, (short)0, acc2, false, false);
            acc3 = __builtin_amdgcn_wmma_f32_16x16x4_f32(false, a, false, b3, (short)0, acc3, false, false);
        }

        // ---- epilogue for the 4 sub-tiles ----
        v8f accs[4] = {acc0, acc1, acc2, acc3};
#pragma unroll
        for (int st = 0; st < 4; ++st) {
            const int   c0   = cb0 + st * 16;
            const int   ccol = c0 + l15;            // this lane's global column
            const int   clab = labels[ccol];
            const float csq  = sq[ccol];
            const v8f   acc  = accs[st];
            if (c0 == i0) {
                // Rare sub-tile containing the diagonal for this wave's rows.
#pragma unroll
                for (int p = 0; p < 8; ++p) {
                    const float t    = __builtin_fmaf(-2.0f, acc[p], csq);
                    const bool  same = (rowlab[p] == clab);
                    const bool  diag = (i0 + p + half * 8 == ccol);
                    hp[p] = (same && !diag) ? fmaxf(hp[p], t) : hp[p];
                    hn[p] = (!same)         ? fminf(hn[p], t) : hn[p];
                }
            } else {
#pragma unroll
                for (int p = 0; p < 8; ++p) {
                    const float t    = __builtin_fmaf(-2.0f, acc[p], csq);
                    const bool  same = (rowlab[p] == clab);
                    hp[p] = same ? fmaxf(hp[p], t) : hp[p];
                    hn[p] = same ? hn[p] : fminf(hn[p], t);
                }
            }
        }
    }

    // Reduce across the 16 column-lanes of each half-wave (xor < 16 stays in half).
#pragma unroll
    for (int p = 0; p < 8; ++p) {
#pragma unroll
        for (int m = 8; m >= 1; m >>= 1) {
            hp[p] = fmaxf(hp[p], __shfl_xor(hp[p], m, 32));
            hn[p] = fminf(hn[p], __shfl_xor(hn[p], m, 32));
        }
    }
    if (l15 == 0) {
#pragma unroll
        for (int p = 0; p < 8; ++p) {
            const int r = i0 + p + half * 8;
            // add the row-norm constant here; +/-inf sentinels survive
            hp_part[cs * N_ROWS + r] = sq[r] + hp[p];
            hn_part[cs * N_ROWS + r] = sq[r] + hn[p];
        }
    }
}

// ---------------------------------------------------------------------------
// Kernel 3: finalize — combine splits, clamp/eps/sqrt, hinge, mean. One block.
// ---------------------------------------------------------------------------
__global__ __launch_bounds__(1024) void bh_finalize(const float* __restrict__ hp_part,
                                                    const float* __restrict__ hn_part,
                                                    float* __restrict__ out) {
    __shared__ float ssum[32];
    __shared__ float scnt[32];
    const int lane = threadIdx.x & 31;
    const int wave = threadIdx.x >> 5;

    float sum = 0.0f, cnt = 0.0f;
    for (int r = threadIdx.x; r < N_ROWS; r += 1024) {
        float hp = -__builtin_inff();
        float hn =  __builtin_inff();
#pragma unroll
        for (int cs = 0; cs < CSPLIT; ++cs) {
            hp = fmaxf(hp, hp_part[cs * N_ROWS + r]);
            hn = fminf(hn, hn_part[cs * N_ROWS + r]);
        }
        const bool valid = (hp != -__builtin_inff()) && (hn != __builtin_inff());
        // deferred clamp + eps-gate + sqrt (monotone, so exact vs reference)
        const float sp = fmaxf(hp, 0.0f);
        const float sn = fmaxf(hn, 0.0f);
        const float dp = (sp > 1e-12f) ? sqrtf(sp) : 0.0f;
        const float dn = (sn > 1e-12f) ? sqrtf(sn) : 0.0f;
        if (valid) {
            sum += fmaxf(dp - dn + TMARGIN, 0.0f);
            cnt += 1.0f;
        }
    }
#pragma unroll
    for (int off = 16; off > 0; off >>= 1) {
        sum += __shfl_xor(sum, off, 32);
        cnt += __shfl_xor(cnt, off, 32);
    }
    if (lane == 0) { ssum[wave] = sum; scnt[wave] = cnt; }
    __syncthreads();
    if (threadIdx.x == 0) {
        float ts = 0.0f, tc = 0.0f;
        for (int w = 0; w < 32; ++w) { ts += ssum[w]; tc += scnt[w]; }
        out[0] = (tc > 0.0f) ? (ts / tc) : 0.0f;
    }
}

// ---------------------------------------------------------------------------
extern "C" void kernel_launch(void* const* d_in, const int* in_sizes, int n_in,
                              void* d_out, int out_size, void* d_ws, size_t ws_size,
                              hipStream_t stream) {
    const float* feats  = (const float*)d_in[0];
    const int*   labels = (const int*)d_in[1];
    float* ws = (float*)d_ws;
    float* sq = ws + WS_SQ;
    float* hp = ws + WS_HP;
    float* hn = ws + WS_HN;

    bh_row_sqnorm<<<N_ROWS / 8, 256, 0, stream>>>(feats, sq);
    bh_tile_kernel<<<dim3(N_ROWS / 128, CSPLIT), 256, 0, stream>>>(feats, labels, sq, hp, hn);
    bh_finalize<<<1, 1024, 0, stream>>>(hp, hn, (float*)d_out);
}